// MyRGCNConv_85126251807558
// MI455X (gfx1250) — compile-verified
//
#include <hip/hip_runtime.h>

// RGCN fused aggregate + per-relation GEMM for MI455X (gfx1250, wave32).
//
// out[N,64] = sum_r segment_sum(x[idx] * (etype==r)) @ W[r]
//           = aggr[N, 512] @ Wcat[512, 64]   (Wcat = weights viewed [8*64, 64])
//
// One 128-thread block (4 wave32) owns 16 nodes:
//   Phase 1: per-wave CSR edge walk. Edge metadata (idx/etype) is batch-loaded
//            coalesced (32 edges per load) and broadcast via __shfl
//            (ds_bpermute); per edge the 32 lanes gather x[src] as float2
//            (coalesced 256B) and ds_add_f32 into LDS aggr[16][512+pad].
//   Phase 2: 4 waves x (16x16) output tiles via V_WMMA_F32_16X16X4_F32,
//            A from LDS (ds_load_2addr_b64, bank-conflict-free via +4 pad),
//            B from global weights (128 KB total, L2-hot across all blocks).

typedef float v2f __attribute__((ext_vector_type(2)));
typedef float v8f __attribute__((ext_vector_type(8)));

#define IN_C  64
#define HID_C 64
#define NUM_REL 8
#define KDIM (NUM_REL * IN_C)     // 512 = reduction length of fused GEMM
#define ROWPAD 4                  // break 512-dword bank aliasing (516%64==4)
#define ROW (KDIM + ROWPAD)       // LDS row stride in floats
#define NODES_PER_BLOCK 16
#define THREADS 128               // 4 wave32

__global__ __launch_bounds__(THREADS)
void rgcn_fused_wmma_kernel(const float* __restrict__ x,      // [N, 64]
                            const float* __restrict__ wgt,    // [8,64,64] == Wcat[512,64]
                            const int*   __restrict__ ptr,    // [N+1]
                            const int*   __restrict__ idx,    // [E]
                            const int*   __restrict__ etype,  // [E]
                            float*       __restrict__ out,    // [N, 64]
                            int num_node)
{
    __shared__ float aggr[NODES_PER_BLOCK * ROW];  // 16 * 516 * 4B = 33 KB

    const int tid  = (int)threadIdx.x;
    const int lane = tid & 31;
    const int wave = tid >> 5;
    const int nodeBase = (int)blockIdx.x * NODES_PER_BLOCK;

    // ---- zero the LDS accumulator tile ----
    for (int i = tid; i < NODES_PER_BLOCK * ROW; i += THREADS)
        aggr[i] = 0.0f;
    __syncthreads();

    // ---- Phase 1: per-relation aggregation into LDS ----
    // Wave w owns local nodes 4w..4w+3 (exclusive writer => no cross-wave
    // race; edges of one node processed sequentially by one wave => no
    // intra-node race, so plain ds float atomics fully resolve ordering).
    for (int t = 0; t < 4; ++t) {
        const int local = wave * 4 + t;
        const int node  = nodeBase + local;
        if (node < num_node) {                    // wave-uniform branch
            const int e0 = ptr[node];
            const int e1 = ptr[node + 1];
            float* rowp = &aggr[local * ROW];

            for (int eb = e0; eb < e1; eb += 32) {
                const int cnt = min(32, e1 - eb); // wave-uniform
                // Coalesced metadata batch: lane l holds edge eb+l.
                int bsrc = 0, brel = 0;
                if (lane < cnt) {
                    bsrc = idx[eb + lane];
                    brel = etype[eb + lane];
                }
                // Uniform loop over the batch; EXEC all-1s again here.
                for (int j = 0; j < cnt; ++j) {
                    const int src = __shfl(bsrc, j, 32);  // ds_bpermute_b32
                    const int rel = __shfl(brel, j, 32);
                    // 32 lanes cover the 64 channels as float2: one
                    // global_load_b64 per lane, coalesced 256B per edge
                    // (x is 12.8 MB -> L2-resident).
                    const v2f v = *(const v2f*)(x + (size_t)src * IN_C + 2 * lane);
                    float* dst = rowp + rel * IN_C + 2 * lane;
                    // ds_add_f32: dwords {2l, 2l+1} -> distinct banks.
                    atomicAdd(dst,     v.x);
                    atomicAdd(dst + 1, v.y);
                }
            }
        }
    }
    __syncthreads();

    // ---- Phase 2: out[16 x 64] tile = aggr[16 x 512] @ Wcat[512 x 64] ----
    // Wave w computes output columns [16w, 16w+16). EXEC is all-1s through
    // the WMMA loop (uniform control flow) as the ISA requires.
    //
    // Fragment layouts (ISA 7.12.2, 32-bit operands):
    //   A(16x4):  lane L -> m = L%16, holds k = 2*(L/16) + j in vgpr j
    //   B(4x16):  lane L -> n = L%16, holds k = 2*(L/16) + j in vgpr j
    //   C/D(16x16): vgpr i -> row m = i + 8*(L/16), col n = L%16
    const int h       = lane >> 4;   // lane-half: selects K pair / row half
    const int mn      = lane & 15;   // A-row (m) and B/C-column (n)
    const int colBase = wave * 16;

    v8f c = {};
    for (int k0 = 0; k0 < KDIM; k0 += 4) {
        // A fragment from LDS: 8-byte aligned ds_load_b64; +4 row pad makes
        // lanes m=0..15 hit banks {4m,4m+1} / {4m+2,4m+3} -> conflict-free.
        const v2f a = *(const v2f*)(&aggr[mn * ROW + k0 + 2 * h]);

        // B fragment from global weights (hot in L1/L2; shared by all blocks).
        v2f b;
        b.x = wgt[(size_t)(k0 + 2 * h + 0) * HID_C + colBase + mn];
        b.y = wgt[(size_t)(k0 + 2 * h + 1) * HID_C + colBase + mn];

        // D = A x B + C, full fp32 (matches reference precision).
        c = __builtin_amdgcn_wmma_f32_16x16x4_f32(
                /*neg_a=*/false, a, /*neg_b=*/false, b,
                /*c_mod=*/(short)0, c, /*reuse_a=*/false, /*reuse_b=*/false);
    }

    // ---- store D fragment ----
    if (nodeBase + NODES_PER_BLOCK <= num_node) {
        // Full tile (always true when N % 16 == 0): unguarded stores with
        // immediate row offsets, no per-element exec juggling.
        float* o = out + (size_t)(nodeBase + 8 * h) * HID_C + colBase + mn;
        #pragma unroll
        for (int i = 0; i < 8; ++i)
            o[(size_t)i * HID_C] = c[i];
    } else {
        #pragma unroll
        for (int i = 0; i < 8; ++i) {
            const int node = nodeBase + i + 8 * h;
            if (node < num_node)
                out[(size_t)node * HID_C + colBase + mn] = c[i];
        }
    }
}

extern "C" void kernel_launch(void* const* d_in, const int* in_sizes, int n_in,
                              void* d_out, int out_size, void* d_ws, size_t ws_size,
                              hipStream_t stream) {
    // setup_inputs() order: x, weights, ptr, idx, edge_types, num_node
    const float* x     = (const float*)d_in[0];
    const float* wgt   = (const float*)d_in[1];
    const int*   ptr   = (const int*)d_in[2];
    const int*   idx   = (const int*)d_in[3];
    const int*   etype = (const int*)d_in[4];
    (void)d_ws; (void)ws_size; (void)n_in; (void)out_size;

    const int num_node = in_sizes[0] / IN_C;   // x is [N, 64]
    const int blocks = (num_node + NODES_PER_BLOCK - 1) / NODES_PER_BLOCK;

    rgcn_fused_wmma_kernel<<<blocks, THREADS, 0, stream>>>(
        x, wgt, ptr, idx, etype, (float*)d_out, num_node);
}